// Loreclayer_67353677136187
// MI455X (gfx1250) — compile-verified
//
#include <hip/hip_runtime.h>
#include <hip/hip_bf16.h>
#include <cstdint>

typedef __attribute__((ext_vector_type(2))) float v2f;
typedef __attribute__((ext_vector_type(8))) float v8f;
typedef __attribute__((ext_vector_type(4))) unsigned int u32x4;
typedef __attribute__((ext_vector_type(8))) int i32x8;
typedef __attribute__((ext_vector_type(4))) int i32x4;

#define BS      8
#define SEQ     2048
#define DIM     2048
#define RNK     8
#define CTR     32
#define HID     60
#define NEXP    4
#define SCALING 2.0f      // 16.0 / r
#define LN_EPS  1e-5f

#define TOK_PER_BLK 64    // tokens per block (4 WMMA tiles of 16)
#define KC      64        // K chunk DMAed into LDS per step (pad_interval=64 DW)
#define XS_STR  (KC + 4)  // TDM pad_amount = 4 DWORDs -> conflict-free 8B gathers
#define AS_STR  (KC + 4)
#define XA_STR  10        // pad 2 -> conflict-free xa gathers
#define NCHUNK  (DIM / KC)

// ---------------------------------------------------------------------------
// Kernel 1: gating MLP. One lane per batch sample (bs = 8).
// ---------------------------------------------------------------------------
__global__ void gate_kernel(const float* __restrict__ ctr,
                            const float* __restrict__ gamma,
                            const float* __restrict__ beta,
                            const float* __restrict__ W1, const float* __restrict__ b1,
                            const float* __restrict__ W2, const float* __restrict__ b2,
                            float* __restrict__ gate_out) {
    int b = threadIdx.x;
    if (b >= BS) return;
    const float* c = ctr + b * CTR;
    float mu = 0.f;
    for (int d = 0; d < CTR; ++d) mu += c[d];
    mu *= (1.f / CTR);
    float var = 0.f;
    for (int d = 0; d < CTR; ++d) { float t = c[d] - mu; var += t * t; }
    var *= (1.f / CTR);
    float inv = rsqrtf(var + LN_EPS);
    float z[CTR];
    for (int d = 0; d < CTR; ++d) z[d] = (c[d] - mu) * inv * gamma[d] + beta[d];
    float h[HID];
    for (int i = 0; i < HID; ++i) {
        float s = b1[i];
        for (int d = 0; d < CTR; ++d) s += z[d] * W1[i * CTR + d];
        h[i] = s > 0.f ? s : 0.f;
    }
    float g[NEXP];
    float mx = -1e30f;
    for (int o = 0; o < NEXP; ++o) {
        float s = b2[o];
        for (int i = 0; i < HID; ++i) s += h[i] * W2[o * HID + i];
        g[o] = s;
        mx = fmaxf(mx, s);
    }
    float se = 0.f;
    for (int o = 0; o < NEXP; ++o) { g[o] = __expf(g[o] - mx); se += g[o]; }
    float rs = 1.f / se;
    for (int o = 0; o < NEXP; ++o) gate_out[b * NEXP + o] = g[o] * rs;
}

// ---------------------------------------------------------------------------
// Kernel 2: A = gate @ Wa^T, B = gate @ Wb^T (scaling folded into B).
// ---------------------------------------------------------------------------
__global__ void ab_kernel(const float* __restrict__ Wa,
                          const float* __restrict__ Wb,
                          const float* __restrict__ gate,
                          float* __restrict__ A_ws,
                          float* __restrict__ B_ws) {
    int idx = blockIdx.x * blockDim.x + threadIdx.x;   // 0 .. 2*131072-1
    int f = idx & (BS * RNK * DIM - 1);
    int b = f >> 14;                                    // f / (RNK*DIM)
    int i = f & (RNK * DIM - 1);
    float4 gv = *(const float4*)(gate + b * 4);
    if (idx < BS * RNK * DIM) {
        float4 wr = *(const float4*)(Wa + (size_t)i * 4);
        A_ws[f] = wr.x * gv.x + wr.y * gv.y + wr.z * gv.z + wr.w * gv.w;
    } else {
        float4 wr = *(const float4*)(Wb + (size_t)i * 4);
        B_ws[f] = (wr.x * gv.x + wr.y * gv.y + wr.z * gv.z + wr.w * gv.w) * SCALING;
    }
}

// ---------------------------------------------------------------------------
// TDM descriptor build + issue: 2D tile (tile_w x tile_h) of fp32, row stride
// `stride_elems`, global -> LDS, with LDS padding of 4 DWORDs after every 64
// DWORDs (pad_interval code 5, pad_amount code 3) => LDS row stride = KC + 4.
// ---------------------------------------------------------------------------
__device__ __forceinline__ void tdm_load_2d(unsigned lds_byte_addr,
                                            const float* gaddr,
                                            unsigned tile_w, unsigned tile_h,
                                            unsigned stride_elems) {
    unsigned long long ga = (unsigned long long)(uintptr_t)gaddr;
    u32x4 g0;
    g0[0] = 1u;                                            // count=1, user-mode
    g0[1] = lds_byte_addr;                                 // LDS byte address
    g0[2] = (unsigned)(ga & 0xffffffffu);                  // global_addr[31:0]
    g0[3] = (unsigned)((ga >> 32) & 0x1ffffffu)            // global_addr[56:32]
          | (2u << 30);                                    // type = 2 (image)
    i32x8 g1;
    g1[0] = (int)((2u << 16)      // data_size = 2 (4 bytes)
          | (1u << 20)            // pad_enable
          | (5u << 22)            // pad_interval code 5 -> every 64 DWORDs
          | (3u << 25));          // pad_amount  code 3 -> 4 DWORDs
    g1[1] = (int)((2048u & 0xffffu) << 16);                // tensor_dim0 lo16
    g1[2] = (int)((2048u >> 16) | (0xffffu << 16));        // dim0 hi | dim1 lo
    g1[3] = (int)(tile_w << 16);                           // dim1 hi | tile_dim0
    g1[4] = (int)(tile_h & 0xffffu);                       // tile_dim1 (dim2=0)
    g1[5] = (int)stride_elems;                             // dim0_stride[31:0]
    g1[6] = 0;                                             // stride hi | dim1_stride lo
    g1[7] = 0;
    i32x4 z4 = {0, 0, 0, 0};
#if __clang_major__ >= 23
    i32x8 z8 = {0, 0, 0, 0, 0, 0, 0, 0};
    __builtin_amdgcn_tensor_load_to_lds(g0, g1, z4, z4, z8, 0);
#else
    __builtin_amdgcn_tensor_load_to_lds(g0, g1, z4, z4, 0);
#endif
}

__device__ __forceinline__ unsigned lds_off(const void* p) {
    return (unsigned)(uintptr_t)p;   // low 32 bits of flat LDS addr = LDS offset
}

// ---------------------------------------------------------------------------
// Kernel 3: out[b] = (x[b] @ A[b]^T) @ B[b].  Both GEMMs on
// V_WMMA_F32_16X16X4_F32; x/A staged global->LDS by the Tensor Data Mover,
// double-buffered so the DMA of chunk i+1 overlaps WMMA compute on chunk i.
// ---------------------------------------------------------------------------
__launch_bounds__(256)
__global__ void lora_main_kernel(const float* __restrict__ x,
                                 const float* __restrict__ A_ws,
                                 const float* __restrict__ B_ws,
                                 float* __restrict__ out) {
    __shared__ __align__(16) float xs[2][TOK_PER_BLK * XS_STR];  // 2 x 17 KB
    __shared__ __align__(16) float as_[2][16 * AS_STR];          // 2 x 4.3 KB
    __shared__ float xa[TOK_PER_BLK * XA_STR];                   // 2.5 KB

    const int b    = blockIdx.y;
    const int s0   = blockIdx.x * TOK_PER_BLK;
    const int tid  = threadIdx.x;
    const int w    = tid >> 5;        // wave id (wave32)
    const int lane = tid & 31;
    const int m    = lane & 15;       // row/col index within 16
    const int g    = lane >> 4;       // half-wave group

    const float* xb = x    + ((size_t)b * SEQ + s0) * DIM;
    const float* Ab = A_ws + (size_t)b * (RNK * DIM);
    const float* Bb = B_ws + (size_t)b * (RNK * DIM);
    float*       ob = out  + ((size_t)b * SEQ + s0) * DIM;

    // Zero N-pad rows (8..15) of both A-chunk buffers once; TDM only writes
    // rows 0..7, so the WMMA B-operand needs no branch for the padded N half.
    for (int v = tid; v < 8 * AS_STR; v += 256) {
        as_[0][8 * AS_STR + v] = 0.f;
        as_[1][8 * AS_STR + v] = 0.f;
    }

    // ---------------- Phase 1: xa = x_tile @ A^T (TDM + WMMA) -------------
    v8f c = {0.f, 0.f, 0.f, 0.f, 0.f, 0.f, 0.f, 0.f};
    const int tt = w & 3;    // token tile owned by this wave
    const int kh = w >> 2;   // K half of each chunk (0: k<32, 1: k>=32)

    if (w == 0) {            // wave-uniform branch: TDM ignores EXEC
        tdm_load_2d(lds_off(&xs[0][0]),  xb, KC, TOK_PER_BLK, DIM);
        tdm_load_2d(lds_off(&as_[0][0]), Ab, KC, RNK,         DIM);
    }

    for (int ci = 0; ci < NCHUNK; ++ci) {
        const int cur = ci & 1;
        if (w == 0) {
            if (ci + 1 < NCHUNK) {
                // prefetch next chunk into the other buffer, then wait for
                // the current chunk (in-order TENSORcnt: <=2 retires pair ci)
                tdm_load_2d(lds_off(&xs[1 - cur][0]),  xb + (ci + 1) * KC,
                            KC, TOK_PER_BLK, DIM);
                tdm_load_2d(lds_off(&as_[1 - cur][0]), Ab + (ci + 1) * KC,
                            KC, RNK, DIM);
                __builtin_amdgcn_s_wait_tensorcnt((short)2);
            } else {
                __builtin_amdgcn_s_wait_tensorcnt((short)0);
            }
        }
        __syncthreads();   // chunk ci visible in LDS to all waves

        const float* xp = &xs[cur][(tt * 16 + m) * XS_STR + 2 * g + kh * (KC / 2)];
        const float* ap = &as_[cur][m * AS_STR + 2 * g + kh * (KC / 2)];
#pragma unroll
        for (int k = 0; k < KC / 2; k += 4) {
            v2f av = *(const v2f*)(xp + k);   // A-op: x[m][k0+2g], x[m][k0+2g+1]
            v2f bv = *(const v2f*)(ap + k);   // B-op: A[n][k0+2g], A[n][k0+2g+1]
            c = __builtin_amdgcn_wmma_f32_16x16x4_f32(false, av, false, bv,
                                                      (short)0, c, false, false);
        }
        __syncthreads();   // done reading buffer `cur` (re-DMAed at ci+2)
    }

    // Combine the two K halves through LDS into xa[token][r].
    if (kh == 1 && m < RNK) {
#pragma unroll
        for (int i = 0; i < 8; ++i)
            xa[(tt * 16 + i + 8 * g) * XA_STR + m] = c[i];
    }
    __syncthreads();
    if (kh == 0 && m < RNK) {
#pragma unroll
        for (int i = 0; i < 8; ++i)
            xa[(tt * 16 + i + 8 * g) * XA_STR + m] += c[i];
    }
    __syncthreads();

    // ---------------- Phase 2: out_tile = xa_tile @ B --------------------
    // Per wave: xa A-operand invariant across its 16 n-chunks per tile; B is
    // read from L2 (1 MB resident). Adjacent waves take adjacent 16-column
    // chunks so the 64B half-line stores combine into full cachelines.
#pragma unroll
    for (int t2 = 0; t2 < 4; ++t2) {
        const float* xr = &xa[(t2 * 16 + m) * XA_STR];
        v2f a0 = *(const v2f*)(xr + 2 * g);       // r = 2g, 2g+1
        v2f a1 = *(const v2f*)(xr + 4 + 2 * g);   // r = 4+2g, 5+2g
        for (int j = 0; j < 16; ++j) {
            int nc  = w + 8 * j;
            int col = nc * 16 + m;
            const float* bp = Bb + col;
            v2f b0 = { bp[(size_t)(2 * g) * DIM],     bp[(size_t)(2 * g + 1) * DIM] };
            v2f b1 = { bp[(size_t)(4 + 2 * g) * DIM], bp[(size_t)(5 + 2 * g) * DIM] };
            v8f d = {0.f, 0.f, 0.f, 0.f, 0.f, 0.f, 0.f, 0.f};
            d = __builtin_amdgcn_wmma_f32_16x16x4_f32(false, a0, false, b0,
                                                      (short)0, d, false, false);
            d = __builtin_amdgcn_wmma_f32_16x16x4_f32(false, a1, false, b1,
                                                      (short)0, d, false, false);
            float* op = ob + (size_t)(t2 * 16 + 8 * g) * DIM + col;
#pragma unroll
            for (int i2 = 0; i2 < 8; ++i2) op[(size_t)i2 * DIM] = d[i2];
        }
    }
}

// ---------------------------------------------------------------------------
extern "C" void kernel_launch(void* const* d_in, const int* in_sizes, int n_in,
                              void* d_out, int out_size, void* d_ws, size_t ws_size,
                              hipStream_t stream) {
    const float* x   = (const float*)d_in[0];
    const float* ctr = (const float*)d_in[1];
    const float* gam = (const float*)d_in[2];
    const float* bet = (const float*)d_in[3];
    const float* W1  = (const float*)d_in[4];
    const float* b1  = (const float*)d_in[5];
    const float* W2  = (const float*)d_in[6];
    const float* b2  = (const float*)d_in[7];
    const float* Wa  = (const float*)d_in[8];
    const float* Wb  = (const float*)d_in[9];
    float* out = (float*)d_out;

    float* gate_ws = (float*)d_ws;                       // 32 floats (256 B slot)
    float* A_ws    = (float*)((char*)d_ws + 256);        // 8*8*2048 floats
    float* B_ws    = A_ws + BS * RNK * DIM;              // 8*8*2048 floats

    gate_kernel<<<1, 32, 0, stream>>>(ctr, gam, bet, W1, b1, W2, b2, gate_ws);
    ab_kernel<<<(2 * BS * RNK * DIM) / 256, 256, 0, stream>>>(Wa, Wb, gate_ws, A_ws, B_ws);
    lora_main_kernel<<<dim3(SEQ / TOK_PER_BLK, BS), 256, 0, stream>>>(x, A_ws, B_ws, out);
}